// K_Means_74423193305442
// MI455X (gfx1250) — compile-verified
//
#include <hip/hip_runtime.h>
#include <hip/hip_bf16.h>

typedef float v2f __attribute__((ext_vector_type(2)));
typedef float v4f __attribute__((ext_vector_type(4)));
typedef float v8f __attribute__((ext_vector_type(8)));

#define N_PTS    262144
#define DIM      64
#define KC       256
#define NWG      512
#define THREADS  256
#define WAVES    8
#define ROWS_PER_WG   (N_PTS / NWG)        // 512
#define ROWS_PER_ITER (WAVES * 16)         // 128
#define NITER         (ROWS_PER_WG / ROWS_PER_ITER) // 4
#define C_STRIDE 68                         // 64 + 4 pad -> bank-conflict-free B reads

__global__ __launch_bounds__(THREADS) void kmeans_main(
    const float* __restrict__ X, const float* __restrict__ C,
    float* __restrict__ out, float* __restrict__ gsums, float* __restrict__ gcounts)
{
    __shared__ float sC[KC * C_STRIDE];   // centroids, padded rows (69632 B)
    __shared__ float sC2[KC];             // ||c||^2
    __shared__ float sSums[KC * DIM];     // segment-sum accumulators (65536 B)
    __shared__ float sCounts[KC];

    const int tid  = threadIdx.x;
    const int lane = tid & 31;
    const int wave = tid >> 5;
    const int il   = lane & 15;   // position within 16-lane half
    const int hf   = lane >> 4;   // which half of the wave

    // --- stage centroids into LDS, compute c2, zero accumulators ---
    {
        const float* crow = C + tid * DIM;   // tid == centroid index (256 threads)
        float acc2 = 0.f;
        #pragma unroll
        for (int j = 0; j < DIM; j += 4) {
            v4f v = *(const v4f*)(crow + j);
            *(v4f*)(sC + tid * C_STRIDE + j) = v;
            acc2 += v.x * v.x + v.y * v.y + v.z * v.z + v.w * v.w;
        }
        sC2[tid] = acc2;
        sCounts[tid] = 0.f;
        for (int idx = tid; idx < KC * DIM; idx += THREADS) sSums[idx] = 0.f;
    }
    __syncthreads();

    float localMinSum = 0.f;
    const int wgBase = blockIdx.x * ROWS_PER_WG;

    for (int it = 0; it < NITER; ++it) {
        const int rowB = wgBase + it * ROWS_PER_ITER + wave * 16;

        // --- preload A fragments for the 16-row slab (f32 16x16x4 layout) ---
        // lane holds row (lane&15); VGPR pair = K-dims {d,d+1} (half0) / {d+2,d+3} (half1)
        v2f a[16];
        const float* xrow = X + (size_t)(rowB + il) * DIM + 2 * hf;
        float part = 0.f;
        #pragma unroll
        for (int dc = 0; dc < 16; ++dc) {
            a[dc] = *(const v2f*)(xrow + dc * 4);
            part += a[dc].x * a[dc].x + a[dc].y * a[dc].y;
        }
        // ||x||^2 per row: combine the two halves, then broadcast per C/D row index
        float x2full = part + __shfl_xor(part, 16, 32);
        float x2m[8];
        #pragma unroll
        for (int e = 0; e < 8; ++e)
            x2m[e] = __shfl(x2full, e + 8 * hf, 32);   // row m = e + 8*half

        float bestVal[8];
        int   bestIdx[8];
        #pragma unroll
        for (int e = 0; e < 8; ++e) { bestVal[e] = __builtin_inff(); bestIdx[e] = 0; }

        // --- 16 column tiles of 16 centroids, processed 2 at a time:
        //     two independent WMMA accumulator chains interleaved to fill the XDL pipe ---
        for (int nt = 0; nt < 16; nt += 2) {
            v8f acc0 = {};
            v8f acc1 = {};
            const float* brow0 = sC + ((nt + 0) * 16 + il) * C_STRIDE + 2 * hf;
            const float* brow1 = sC + ((nt + 1) * 16 + il) * C_STRIDE + 2 * hf;
            #pragma unroll
            for (int dc = 0; dc < 16; ++dc) {
                v2f b0 = *(const v2f*)(brow0 + dc * 4);
                v2f b1 = *(const v2f*)(brow1 + dc * 4);
                acc0 = __builtin_amdgcn_wmma_f32_16x16x4_f32(
                    false, a[dc], false, b0, (short)0, acc0, false, false);
                acc1 = __builtin_amdgcn_wmma_f32_16x16x4_f32(
                    false, a[dc], false, b1, (short)0, acc1, false, false);
            }
            const float c2v0  = sC2[(nt + 0) * 16 + il];
            const float c2v1  = sC2[(nt + 1) * 16 + il];
            const int   nIdx0 = (nt + 0) * 16 + il;
            const int   nIdx1 = (nt + 1) * 16 + il;
            #pragma unroll
            for (int e = 0; e < 8; ++e) {
                float d0 = x2m[e] + c2v0 - 2.0f * acc0[e];
                float d1 = x2m[e] + c2v1 - 2.0f * acc1[e];
                if (d0 < bestVal[e]) { bestVal[e] = d0; bestIdx[e] = nIdx0; }
                if (d1 < bestVal[e]) { bestVal[e] = d1; bestIdx[e] = nIdx1; }
            }
        }

        // --- per-row argmin across the 16 lanes of each half; then scatter ---
        #pragma unroll
        for (int e = 0; e < 8; ++e) {
            #pragma unroll
            for (int off = 8; off >= 1; off >>= 1) {
                float ov = __shfl_xor(bestVal[e], off, 32);
                int   oi = __shfl_xor(bestIdx[e], off, 32);
                if (ov < bestVal[e] || (ov == bestVal[e] && oi < bestIdx[e])) {
                    bestVal[e] = ov; bestIdx[e] = oi;
                }
            }
            const int row = rowB + e + 8 * hf;
            if (il == 0) {
                out[row] = (float)bestIdx[e];     // assignment (as f32 output)
                localMinSum += bestVal[e];
            }
            // 16 lanes of this half each add 4 dims of the row into sums[best]
            v4f xv = *(const v4f*)(X + (size_t)row * DIM + 4 * il);
            float* dst = sSums + bestIdx[e] * DIM + 4 * il;
            atomicAdd(dst + 0, xv.x);
            atomicAdd(dst + 1, xv.y);
            atomicAdd(dst + 2, xv.z);
            atomicAdd(dst + 3, xv.w);
            if (il == 0) atomicAdd(&sCounts[bestIdx[e]], 1.0f);
        }
    }
    __syncthreads();

    // --- flush LDS partials to global ---
    for (int idx = tid; idx < KC * DIM; idx += THREADS)
        unsafeAtomicAdd(&gsums[idx], sSums[idx]);
    unsafeAtomicAdd(&gcounts[tid], sCounts[tid]);

    // --- wave-reduce the min-distance partial, one atomic per wave ---
    #pragma unroll
    for (int off = 16; off >= 1; off >>= 1)
        localMinSum += __shfl_xor(localMinSum, off, 32);
    if (lane == 0)
        unsafeAtomicAdd(out + N_PTS + KC * DIM, localMinSum);
}

__global__ void kmeans_finalize(const float* __restrict__ gsums,
                                const float* __restrict__ gcounts,
                                float* __restrict__ out)
{
    int idx = blockIdx.x * blockDim.x + threadIdx.x;   // 0..K*D-1
    int k = idx >> 6;
    out[N_PTS + idx] = gsums[idx] / gcounts[k];
}

extern "C" void kernel_launch(void* const* d_in, const int* in_sizes, int n_in,
                              void* d_out, int out_size, void* d_ws, size_t ws_size,
                              hipStream_t stream) {
    const float* X = (const float*)d_in[0];   // [N, D]
    const float* C = (const float*)d_in[1];   // [K, D]
    float* out     = (float*)d_out;           // [N] assign + [K*D] centroids + [1] sumdist
    float* gsums   = (float*)d_ws;            // [K*D]
    float* gcounts = gsums + KC * DIM;        // [K]

    hipMemsetAsync(d_ws, 0, (KC * DIM + KC) * sizeof(float), stream);
    hipMemsetAsync(out + N_PTS + KC * DIM, 0, sizeof(float), stream);

    kmeans_main<<<NWG, THREADS, 0, stream>>>(X, C, out, gsums, gcounts);
    kmeans_finalize<<<(KC * DIM) / 256, 256, 0, stream>>>(gsums, gcounts, out);
}